// GraphClassifier_73272142070373
// MI455X (gfx1250) — compile-verified
//
#include <hip/hip_runtime.h>

// ---------------------------------------------------------------------------
// GCN graph classifier on MI455X (gfx1250).
// GEMMs use V_WMMA_F32_16X16X4_F32 (fp32 in/out, wave32, 16x16 tile per wave).
// Scatter/pool phases are L2-resident atomic passes (bandwidth bound).
// ---------------------------------------------------------------------------

typedef float v2f __attribute__((ext_vector_type(2)));
typedef float v8f __attribute__((ext_vector_type(8)));

#define HDIM 128
#define OUTC 2

// ------------------------- small utility kernels ---------------------------

__global__ void k_fill(float* __restrict__ p, float v, int n) {
    int i = blockIdx.x * blockDim.x + threadIdx.x;
    if (i < n) p[i] = v;
}

// deg[col] += ew  (deg pre-filled with 1.0 = self-loop weight)
__global__ void k_deg_scatter(const int* __restrict__ col,
                              const float* __restrict__ ew,
                              float* __restrict__ deg, int E) {
    int e = blockIdx.x * blockDim.x + threadIdx.x;
    if (e < E) atomicAdd(&deg[col[e]], ew[e]);
}

// deg -> 1/sqrt(deg) in place
__global__ void k_deg_finalize(float* __restrict__ deg, int n) {
    int i = blockIdx.x * blockDim.x + threadIdx.x;
    if (i < n) {
        float d = deg[i];
        deg[i] = (d > 0.0f) ? (1.0f / sqrtf(d)) : 0.0f;
    }
}

// ------------------------- WMMA fp32 GEMM [M,128]@[128,128] ----------------
// Block = 256 threads (8 waves). Wave w owns N-tile w (n0 = 16*w).
// blockIdx.x owns M-tile (m0 = 16*blockIdx.x). M must be a multiple of 16
// (N = 100000 = 6250 * 16).
__global__ void __launch_bounds__(256)
k_gemm128_wmma(const float* __restrict__ A,   // [M,128] row-major
               const float* __restrict__ W,   // [128,128] row-major
               float* __restrict__ Out) {     // [M,128] row-major
    const int wave  = threadIdx.x >> 5;
    const int lane  = threadIdx.x & 31;
    const int lhalf = lane >> 4;      // 0: lanes 0-15, 1: lanes 16-31
    const int l16   = lane & 15;
    const int m0    = blockIdx.x * 16;
    const int n0    = wave * 16;

    const float* __restrict__ arow = A + (size_t)(m0 + l16) * HDIM;
    v8f c = {};

#pragma unroll
    for (int k = 0; k < HDIM; k += 4) {
        // A (16x4 f32): lane<16 holds K=k,k+1 ; lane>=16 holds K=k+2,k+3
        v2f a;
        a.x = arow[k + 2 * lhalf];
        a.y = arow[k + 2 * lhalf + 1];
        // B (4x16 f32): row K striped across lane halves, N = n0 + l16
        v2f b;
        b.x = W[(size_t)(k + 2 * lhalf) * HDIM + n0 + l16];
        b.y = W[(size_t)(k + 2 * lhalf + 1) * HDIM + n0 + l16];
        c = __builtin_amdgcn_wmma_f32_16x16x4_f32(
                /*neg_a=*/false, a, /*neg_b=*/false, b,
                /*c_mod=*/(short)0, c, /*reuse_a=*/false, /*reuse_b=*/false);
    }

    // C/D layout: VGPR i -> row m0 + i + 8*lhalf, col n0 + l16
#pragma unroll
    for (int i = 0; i < 8; ++i) {
        Out[(size_t)(m0 + i + 8 * lhalf) * HDIM + n0 + l16] = c[i];
    }
}

// ------------------------- aggregation kernels -----------------------------

// agg[n][:] = xw[n][:] * dinv[n]^2   (self-loop term, also zero-initializes)
__global__ void k_selfloop_init(const float* __restrict__ xw,
                                const float* __restrict__ dinv,
                                float* __restrict__ agg, int N) {
    size_t i = (size_t)blockIdx.x * blockDim.x + threadIdx.x;
    size_t tot = (size_t)N * HDIM;
    if (i < tot) {
        float d = dinv[i >> 7];   // node id = i / 128
        agg[i] = xw[i] * d * d;
    }
}

// One wave per edge, 4 floats per lane: agg[col] += xw[row] * norm
__global__ void k_edge_scatter(const float* __restrict__ xw,
                               float* __restrict__ agg,
                               const int* __restrict__ row,
                               const int* __restrict__ col,
                               const float* __restrict__ ew,
                               const float* __restrict__ dinv, int E) {
    long long gid = (long long)blockIdx.x * blockDim.x + threadIdx.x;
    int e    = (int)(gid >> 5);
    int lane = (int)(gid & 31);
    if (e >= E) return;
    int r = row[e];
    int c = col[e];
    float norm = dinv[r] * ew[e] * dinv[c];
    float4 v = ((const float4*)(xw + (size_t)r * HDIM))[lane];
    float* dst = agg + (size_t)c * HDIM + lane * 4;
    atomicAdd(dst + 0, v.x * norm);
    atomicAdd(dst + 1, v.y * norm);
    atomicAdd(dst + 2, v.z * norm);
    atomicAdd(dst + 3, v.w * norm);
}

// h = relu(agg + bias)
__global__ void k_bias_relu(const float* __restrict__ agg,
                            const float* __restrict__ bias,
                            float* __restrict__ h, int N) {
    size_t i = (size_t)blockIdx.x * blockDim.x + threadIdx.x;
    size_t tot = (size_t)N * HDIM;
    if (i < tot) {
        float v = agg[i] + bias[i & (HDIM - 1)];
        h[i] = v > 0.0f ? v : 0.0f;
    }
}

// ------------------------- pooling + head ----------------------------------

__global__ void k_pool_cnt(const int* __restrict__ batch,
                           float* __restrict__ cnt, int N) {
    int i = blockIdx.x * blockDim.x + threadIdx.x;
    if (i < N) atomicAdd(&cnt[batch[i]], 1.0f);
}

// One wave per node, 4 floats per lane: pooled[batch[n]] += h[n]
__global__ void k_pool_sum(const float* __restrict__ h,
                           const int* __restrict__ batch,
                           float* __restrict__ pooled, int N) {
    long long gid = (long long)blockIdx.x * blockDim.x + threadIdx.x;
    int n    = (int)(gid >> 5);
    int lane = (int)(gid & 31);
    if (n >= N) return;
    int g = batch[n];
    float4 v = ((const float4*)(h + (size_t)n * HDIM))[lane];
    float* dst = pooled + (size_t)g * HDIM + lane * 4;
    atomicAdd(dst + 0, v.x);
    atomicAdd(dst + 1, v.y);
    atomicAdd(dst + 2, v.z);
    atomicAdd(dst + 3, v.w);
}

// out[g][o] = (pooled[g]/max(cnt[g],1)) . Wl[:,o] + bl[o]
__global__ void k_head(const float* __restrict__ pooled,
                       const float* __restrict__ cnt,
                       const float* __restrict__ Wl,
                       const float* __restrict__ bl,
                       float* __restrict__ out, int G) {
    int gid = blockIdx.x * blockDim.x + threadIdx.x;
    int g = gid >> 1;
    int o = gid & 1;
    if (g >= G) return;
    float inv = 1.0f / fmaxf(cnt[g], 1.0f);
    const float* pr = pooled + (size_t)g * HDIM;
    float acc = 0.0f;
#pragma unroll 8
    for (int f = 0; f < HDIM; ++f) acc += pr[f] * Wl[f * OUTC + o];
    out[g * OUTC + o] = acc * inv + bl[o];
}

// ------------------------- driver ------------------------------------------

extern "C" void kernel_launch(void* const* d_in, const int* in_sizes, int n_in,
                              void* d_out, int out_size, void* d_ws, size_t ws_size,
                              hipStream_t stream) {
    const float* x     = (const float*)d_in[0];
    const int*   ei    = (const int*)  d_in[1];   // [2,E] flat
    const float* ew    = (const float*)d_in[2];
    const int*   batch = (const int*)  d_in[3];
    const float* W1    = (const float*)d_in[4];
    const float* b1    = (const float*)d_in[5];
    const float* W2    = (const float*)d_in[6];
    const float* b2    = (const float*)d_in[7];
    const float* Wl    = (const float*)d_in[8];
    const float* bl    = (const float*)d_in[9];
    float*       out   = (float*)d_out;

    const int N = in_sizes[0] / HDIM;     // 100000
    const int E = in_sizes[2];            // 1600000
    const int G = out_size / OUTC;        // 512
    const int* row = ei;
    const int* col = ei + E;

    // workspace layout (floats)
    float* ws     = (float*)d_ws;
    size_t NH     = (size_t)N * HDIM;
    float* buf0   = ws;               // xw1 -> h1 -> agg2
    float* buf1   = ws + NH;          // agg1 -> xw2 -> h2
    float* dinv   = ws + 2 * NH;      // N   (deg then 1/sqrt(deg))
    float* pooled = dinv + N;         // G*HDIM
    float* cnt    = pooled + (size_t)G * HDIM;  // G

    const int TB = 256;
    dim3 blk(TB);
    unsigned gN   = (unsigned)((N + TB - 1) / TB);
    unsigned gE   = (unsigned)((E + TB - 1) / TB);
    unsigned gNH  = (unsigned)((NH + TB - 1) / TB);
    unsigned gE32 = (unsigned)(((long long)E * 32 + TB - 1) / TB);
    unsigned gN32 = (unsigned)(((long long)N * 32 + TB - 1) / TB);
    unsigned gGemm = (unsigned)(N / 16);  // N is a multiple of 16

    // 1) symmetric normalization: deg = 1 (self loop) + scatter(ew), dinv = rsqrt
    k_fill<<<gN, blk, 0, stream>>>(dinv, 1.0f, N);
    k_deg_scatter<<<gE, blk, 0, stream>>>(col, ew, dinv, E);
    k_deg_finalize<<<gN, blk, 0, stream>>>(dinv, N);

    // zero pooling accumulators (every call; harness does not re-poison)
    k_fill<<<(unsigned)((G * HDIM + TB - 1) / TB), blk, 0, stream>>>(pooled, 0.0f, G * HDIM);
    k_fill<<<(unsigned)((G + TB - 1) / TB), blk, 0, stream>>>(cnt, 0.0f, G);

    // 2) layer 1: xw1 = x @ W1 ; agg1 = scatter-norm(xw1) ; h1 = relu(agg1+b1)
    k_gemm128_wmma<<<gGemm, blk, 0, stream>>>(x, W1, buf0);
    k_selfloop_init<<<gNH, blk, 0, stream>>>(buf0, dinv, buf1, N);
    k_edge_scatter<<<gE32, blk, 0, stream>>>(buf0, buf1, row, col, ew, dinv, E);
    k_bias_relu<<<gNH, blk, 0, stream>>>(buf1, b1, buf0, N);   // buf0 = h1

    // 3) layer 2: xw2 = h1 @ W2 ; agg2 ; h2 = relu(agg2+b2)
    k_gemm128_wmma<<<gGemm, blk, 0, stream>>>(buf0, W2, buf1); // buf1 = xw2
    k_selfloop_init<<<gNH, blk, 0, stream>>>(buf1, dinv, buf0, N); // buf0 = agg2
    k_edge_scatter<<<gE32, blk, 0, stream>>>(buf1, buf0, row, col, ew, dinv, E);
    k_bias_relu<<<gNH, blk, 0, stream>>>(buf0, b2, buf1, N);   // buf1 = h2

    // 4) global mean pool + linear head
    k_pool_cnt<<<gN, blk, 0, stream>>>(batch, cnt, N);
    k_pool_sum<<<gN32, blk, 0, stream>>>(buf1, batch, pooled, N);
    k_head<<<(unsigned)((G * OUTC + TB - 1) / TB), blk, 0, stream>>>(pooled, cnt, Wl, bl, out, G);
}